// CostVolumeAttention15_66546223284245
// MI455X (gfx1250) — compile-verified
//
#include <hip/hip_runtime.h>
#include <hip/hip_bf16.h>

typedef float v2f __attribute__((ext_vector_type(2)));
typedef float v8f __attribute__((ext_vector_type(8)));

#define TILE_M 64
#define TILE_N 128
#define KC 32
#define LDS_PAD 34  // row stride in dwords; conflict-free for fragment gathers

// ---------------------------------------------------------------------------
// GEMM: Out[M,Nc] = A[M,K] @ Bm[Nc,K]^T (+ bias[Nc])
// fp32 WMMA 16x16x4, wave32. Block = 256 threads = 8 waves, 64x128 tile.
// Each wave: 16 rows x 64 cols (4 N-subtiles) -> 1 A-frag feeds 4 WMMAs.
// Register double-buffering hides global->LDS staging behind the WMMA chain.
// ---------------------------------------------------------------------------
__global__ __launch_bounds__(256) void gemm_rt_wmma(
    const float* __restrict__ A, const float* __restrict__ Bm,
    const float* __restrict__ bias, float* __restrict__ Out,
    int M, int K, int Nc)
{
    __shared__ float As[TILE_M][LDS_PAD];
    __shared__ float Bs[TILE_N][LDS_PAD];

    const int tid  = threadIdx.x;
    const int m0   = blockIdx.x * TILE_M;
    const int n0   = blockIdx.y * TILE_N;
    const int wave = tid >> 5;
    const int lane = tid & 31;
    const int wm   = wave & 3;   // 4 waves along M
    const int wn   = wave >> 2;  // 2 waves along N, 64 cols each
    const int lm   = lane & 15;
    const int kh   = lane >> 4;  // 0/1 half of wave

    // staging registers: 2 float4 for A tile, 4 float4 for B tile
    float4 ra[2], rb[4];
    // per-thread staging coordinates (row, kcol) for the 6 float4 pieces
    int arow[2], acol[2], brow[4], bcol[4];
    #pragma unroll
    for (int j = 0; j < 2; ++j) {
        int li = tid + j * 256;          // 0..511 -> A tile (64x32 = 512 float4)
        arow[j] = li >> 3;
        acol[j] = (li & 7) << 2;
    }
    #pragma unroll
    for (int j = 0; j < 4; ++j) {
        int li = tid + j * 256;          // 0..1023 -> B tile (128x32 = 1024 float4)
        brow[j] = li >> 3;
        bcol[j] = (li & 7) << 2;
    }

    v8f c[4];
    #pragma unroll
    for (int s = 0; s < 4; ++s) c[s] = (v8f){};

    // prologue: stage chunk 0
    #pragma unroll
    for (int j = 0; j < 2; ++j)
        ra[j] = *(const float4*)(A + (size_t)(m0 + arow[j]) * K + acol[j]);
    #pragma unroll
    for (int j = 0; j < 4; ++j)
        rb[j] = *(const float4*)(Bm + (size_t)(n0 + brow[j]) * K + bcol[j]);
    #pragma unroll
    for (int j = 0; j < 2; ++j) {
        As[arow[j]][acol[j]+0] = ra[j].x; As[arow[j]][acol[j]+1] = ra[j].y;
        As[arow[j]][acol[j]+2] = ra[j].z; As[arow[j]][acol[j]+3] = ra[j].w;
    }
    #pragma unroll
    for (int j = 0; j < 4; ++j) {
        Bs[brow[j]][bcol[j]+0] = rb[j].x; Bs[brow[j]][bcol[j]+1] = rb[j].y;
        Bs[brow[j]][bcol[j]+2] = rb[j].z; Bs[brow[j]][bcol[j]+3] = rb[j].w;
    }
    __syncthreads();

    for (int k0 = 0; k0 < K; k0 += KC) {
        const bool last = (k0 + KC >= K);
        if (!last) {
            // prefetch next chunk into registers (overlaps WMMA chain)
            #pragma unroll
            for (int j = 0; j < 2; ++j)
                ra[j] = *(const float4*)(A + (size_t)(m0 + arow[j]) * K + k0 + KC + acol[j]);
            #pragma unroll
            for (int j = 0; j < 4; ++j)
                rb[j] = *(const float4*)(Bm + (size_t)(n0 + brow[j]) * K + k0 + KC + bcol[j]);
        }

        #pragma unroll
        for (int kk = 0; kk < KC; kk += 4) {
            // A frag: value j -> A[m][kk + j + 2*kh]; B frag mirrors in K
            v2f a = *(const v2f*)&As[wm * 16 + lm][kk + 2 * kh];
            #pragma unroll
            for (int s = 0; s < 4; ++s) {
                v2f b = *(const v2f*)&Bs[wn * 64 + s * 16 + lm][kk + 2 * kh];
                c[s] = __builtin_amdgcn_wmma_f32_16x16x4_f32(
                        false, a, false, b, (short)0, c[s], false, false);
            }
        }
        __syncthreads();   // all LDS reads of this chunk done

        if (!last) {
            // commit prefetched chunk
            #pragma unroll
            for (int j = 0; j < 2; ++j) {
                As[arow[j]][acol[j]+0] = ra[j].x; As[arow[j]][acol[j]+1] = ra[j].y;
                As[arow[j]][acol[j]+2] = ra[j].z; As[arow[j]][acol[j]+3] = ra[j].w;
            }
            #pragma unroll
            for (int j = 0; j < 4; ++j) {
                Bs[brow[j]][bcol[j]+0] = rb[j].x; Bs[brow[j]][bcol[j]+1] = rb[j].y;
                Bs[brow[j]][bcol[j]+2] = rb[j].z; Bs[brow[j]][bcol[j]+3] = rb[j].w;
            }
            __syncthreads();
        }
    }

    // Epilogue: C layout — VGPR v, half kh: row = v + 8*kh, col = lm
    #pragma unroll
    for (int s = 0; s < 4; ++s) {
        int col = n0 + wn * 64 + s * 16 + lm;
        float bv = bias ? bias[col] : 0.0f;
        #pragma unroll
        for (int v = 0; v < 8; ++v) {
            int row = m0 + wm * 16 + v + 8 * kh;
            Out[(size_t)row * Nc + col] = c[s][v] + bv;
        }
    }
}

// ---------------------------------------------------------------------------
// Reshape qk [B,N,1024] -> channel-major vid layout [128][64][1024],
// with per-pixel L2 normalization over the 64 channels and k time-shift.
// ---------------------------------------------------------------------------
__global__ __launch_bounds__(256) void reshape_norm(
    const float* __restrict__ qk, float* __restrict__ qvn, float* __restrict__ kvn)
{
    const int f = blockIdx.x;                       // frame 0..127
    const int p = blockIdx.y * 256 + threadIdx.x;   // pixel 0..1023
    const int b = f >> 6;
    const int h = (f >> 3) & 7;
    const int t = f & 7;
    const size_t fbase = (size_t)f * 65536;

    // ---- q ----
    {
        int n = t * 1024 + p;
        const float4* src = (const float4*)(qk + ((size_t)(b * 8192 + n)) * 1024 + h * 64);
        float4 v[16];
        float ss = 0.0f;
        #pragma unroll
        for (int i = 0; i < 16; ++i) {
            v[i] = src[i];
            ss += v[i].x * v[i].x + v[i].y * v[i].y + v[i].z * v[i].z + v[i].w * v[i].w;
        }
        float s = 1.0f / fmaxf(sqrtf(ss), 1e-12f);
        #pragma unroll
        for (int i = 0; i < 16; ++i) {
            qvn[fbase + (size_t)(4 * i + 0) * 1024 + p] = v[i].x * s;
            qvn[fbase + (size_t)(4 * i + 1) * 1024 + p] = v[i].y * s;
            qvn[fbase + (size_t)(4 * i + 2) * 1024 + p] = v[i].z * s;
            qvn[fbase + (size_t)(4 * i + 3) * 1024 + p] = v[i].w * s;
        }
    }
    // ---- k (time-shifted forward, repeat last frame) ----
    {
        int ts = (t == 7) ? 7 : t + 1;
        int n = ts * 1024 + p;
        const float4* src = (const float4*)(qk + ((size_t)(b * 8192 + n)) * 1024 + 512 + h * 64);
        float4 v[16];
        float ss = 0.0f;
        #pragma unroll
        for (int i = 0; i < 16; ++i) {
            v[i] = src[i];
            ss += v[i].x * v[i].x + v[i].y * v[i].y + v[i].z * v[i].z + v[i].w * v[i].w;
        }
        float s = 1.0f / fmaxf(sqrtf(ss), 1e-12f);
        #pragma unroll
        for (int i = 0; i < 16; ++i) {
            kvn[fbase + (size_t)(4 * i + 0) * 1024 + p] = v[i].x * s;
            kvn[fbase + (size_t)(4 * i + 1) * 1024 + p] = v[i].y * s;
            kvn[fbase + (size_t)(4 * i + 2) * 1024 + p] = v[i].z * s;
            kvn[fbase + (size_t)(4 * i + 3) * 1024 + p] = v[i].w * s;
        }
    }
}

// ---------------------------------------------------------------------------
// 7x7 local cost volume (zero-padded) + 49->64 projection, fused.
// Block = (frame, 4-row strip), 128 threads = 1 pixel each.
// Output written directly in final [B, N, C] layout (contiguous 64 floats).
// ---------------------------------------------------------------------------
#define CHC 16
__global__ __launch_bounds__(128) void costvol_proj(
    const float* __restrict__ qvn, const float* __restrict__ kvn,
    const float* __restrict__ w_corr, const float* __restrict__ b_corr,
    float* __restrict__ mf)
{
    __shared__ float Ks[CHC][10][40];   // 16ch x (4+6 halo rows) x padded cols
    __shared__ float Wc[64 * 49];
    __shared__ float Bc[64];

    const int f   = blockIdx.x;
    const int y0  = blockIdx.y * 4;
    const int tid = threadIdx.x;
    const int x   = tid & 31;
    const int yl  = tid >> 5;           // 0..3
    const int y   = y0 + yl;
    const size_t fbase = (size_t)f * 65536;

    for (int i = tid; i < 64 * 49; i += 128) Wc[i] = w_corr[i];
    if (tid < 64) Bc[tid] = b_corr[tid];

    float corr[49];
    #pragma unroll
    for (int i = 0; i < 49; ++i) corr[i] = 0.0f;

    for (int c0 = 0; c0 < 64; c0 += CHC) {
        __syncthreads();  // previous Ks reads done (also covers Wc/Bc writes)
        for (int i = tid; i < CHC * 10 * 40; i += 128) {
            int lc = i % 40;
            int r  = (i / 40) % 10;
            int ch = i / 400;
            int gx = lc - 3;
            int gy = y0 + r - 3;
            float val = 0.0f;
            if (gx >= 0 && gx < 32 && gy >= 0 && gy < 32)
                val = kvn[fbase + (size_t)(c0 + ch) * 1024 + gy * 32 + gx];
            Ks[ch][r][lc] = val;
        }
        __syncthreads();

        float qr[CHC];
        #pragma unroll
        for (int ch = 0; ch < CHC; ++ch)
            qr[ch] = qvn[fbase + (size_t)(c0 + ch) * 1024 + y * 32 + x];

        #pragma unroll
        for (int u = 0; u < 7; ++u) {
            #pragma unroll
            for (int v = 0; v < 7; ++v) {
                float acc = corr[u * 7 + v];
                #pragma unroll
                for (int ch = 0; ch < CHC; ++ch)
                    acc += qr[ch] * Ks[ch][yl + u][x + v];
                corr[u * 7 + v] = acc;
            }
        }
    }
    __syncthreads();

    // projection 49 -> 64, write into [B, N, C] layout
    const int b = f >> 6;
    const int h = (f >> 3) & 7;
    const int t = f & 7;
    const int n = t * 1024 + y * 32 + x;
    float* outp = mf + ((size_t)(b * 8192 + n)) * 512 + h * 64;

    #pragma unroll 4
    for (int g = 0; g < 16; ++g) {
        float4 r;
        float* rp = (float*)&r;
        #pragma unroll
        for (int j = 0; j < 4; ++j) {
            int dch = g * 4 + j;
            float acc = Bc[dch];
            const float* w = &Wc[dch * 49];
            #pragma unroll
            for (int uv = 0; uv < 49; ++uv) acc += w[uv] * corr[uv];
            rp[j] = acc;
        }
        ((float4*)outp)[g] = r;
    }
}

// ---------------------------------------------------------------------------
extern "C" void kernel_launch(void* const* d_in, const int* in_sizes, int n_in,
                              void* d_out, int out_size, void* d_ws, size_t ws_size,
                              hipStream_t stream)
{
    const float* x      = (const float*)d_in[0];  // [2, 8192, 512]
    const float* w_qk   = (const float*)d_in[1];  // [1024, 512]
    const float* w_corr = (const float*)d_in[2];  // [64, 49]
    const float* b_corr = (const float*)d_in[3];  // [64]
    const float* w_proj = (const float*)d_in[4];  // [512, 512]
    const float* b_proj = (const float*)d_in[5];  // [512]
    float* out = (float*)d_out;                   // [2, 8192, 512]

    const int M = 16384, K = 512;

    float* qk  = (float*)d_ws;                      // 16384*1024 floats (64 MB)
    float* qvn = qk + (size_t)16384 * 1024;         // 128*64*1024 floats (32 MB)
    float* kvn = qvn + (size_t)128 * 65536;         // 32 MB
    float* mf  = qk;                                // reuse dead qk region (32 MB)

    // 1) qk = x @ w_qk^T
    gemm_rt_wmma<<<dim3(M / TILE_M, 1024 / TILE_N), 256, 0, stream>>>(
        x, w_qk, nullptr, qk, M, K, 1024);

    // 2) split q/k, time-shift k, L2-normalize, channel-major vid layout
    reshape_norm<<<dim3(128, 4), 256, 0, stream>>>(qk, qvn, kvn);

    // 3) cost volume + 49->64 projection, fused, output in [B,N,C] layout
    costvol_proj<<<dim3(128, 8), 128, 0, stream>>>(qvn, kvn, w_corr, b_corr, mf);

    // 4) out = mf @ w_proj^T + b_proj
    gemm_rt_wmma<<<dim3(M / TILE_M, 512 / TILE_N), 256, 0, stream>>>(
        mf, w_proj, b_proj, out, M, K, 512);
}